// tensor_GCN_69604239999340
// MI455X (gfx1250) — compile-verified
//
#include <hip/hip_runtime.h>

// Problem constants (match reference)
#define BB   128
#define WW   4
#define NNN  512
#define FIN  64
#define HH1  32
#define HH2  32
#define PSTR 80    // dword stride per K-pair row in paired sup layout (pad for bank spread)

typedef float v2f __attribute__((ext_vector_type(2)));
typedef float v4f __attribute__((ext_vector_type(4)));
typedef float v8f __attribute__((ext_vector_type(8)));

// Full-precision fp32 WMMA: D(16x16) = A(16x4) * B(4x16) + C
__device__ __forceinline__ v8f wmma4(v2f a, v2f b, v8f c) {
  return __builtin_amdgcn_wmma_f32_16x16x4_f32(
      /*neg_a=*/false, a, /*neg_b=*/false, b,
      /*c_mod=*/(short)0, c, /*reuse_a=*/false, /*reuse_b=*/false);
}

// ---------------------------------------------------------------------------
// Kernel 1: per (b,w) block.
//   Phase 1: sup = x[b,w] (512x64) @ W_intra[w] (64x32)  -> LDS (paired layout)
//   Phase 2: outs[b,w] = adj[b,w] (512x512) @ sup (512x32) -> workspace
//
// A fragments use a consistent K-slot permutation so each lane does one b128
// load covering two WMMA K-steps; B fragments use the same mapping:
//   step1 hw K-slots = phys {k,k+1 (lanes 0-15), k+4,k+5 (lanes 16-31)}
//   step2 hw K-slots = phys {k+2,k+3,            k+6,k+7}
//
// sup is stored K-paired+padded: supP[(k/2)*PSTR + n*2 + (k&1)], so each B
// fragment is ONE aligned ds_load_b64, and the +2-pair offset of the hi lane
// half (160 dwords ≡ 32 mod 64 banks) makes every fragment load conflict-free.
// ---------------------------------------------------------------------------
__global__ __launch_bounds__(256)
void gcn_intra_kernel(const float* __restrict__ x,
                      const float* __restrict__ adj,
                      const float* __restrict__ Wintra,
                      float* __restrict__ outs)
{
  __shared__ float sW[FIN * HH1];          // 8 KB:  W_intra[w]
  __shared__ float supP[(NNN / 2) * PSTR]; // 80 KB: support, K-paired layout

  const int bw   = blockIdx.x;             // b*W + w
  const int w    = bw & (WW - 1);
  const int tid  = threadIdx.x;
  const int wave = tid >> 5;               // 0..7
  const int lane = tid & 31;
  const int lr   = lane & 15;              // row-in-tile (A) / col-in-tile (B,C,D)
  const int lhi  = lane >> 4;              // K-half select

  // Load W_intra[w] into LDS
  const float* Wg = Wintra + (size_t)w * FIN * HH1;
  for (int i = tid; i < FIN * HH1; i += 256) sW[i] = Wg[i];
  __syncthreads();

  // ---- Phase 1: support into LDS (paired layout) via WMMA ----
  const float* xg = x + (size_t)bw * NNN * FIN;
  for (int mi = 0; mi < 4; ++mi) {
    const int mt  = wave * 4 + mi;         // 32 M-tiles over 8 waves
    const int row = mt * 16 + lr;
    const float* ap = xg + (size_t)row * FIN + lhi * 4;
    v8f acc0 = {}; v8f acc1 = {};
    #pragma unroll
    for (int k = 0; k < FIN; k += 8) {
      v4f a4 = *(const v4f*)(ap + k);
      v2f aA = {a4.x, a4.y};
      v2f aB = {a4.z, a4.w};
      const int r1 = (k + lhi * 4) * HH1;
      const int r2 = r1 + 2 * HH1;
      v2f b00 = { sW[r1 + lr],      sW[r1 + HH1 + lr] };
      v2f b01 = { sW[r1 + 16 + lr], sW[r1 + HH1 + 16 + lr] };
      v2f b10 = { sW[r2 + lr],      sW[r2 + HH1 + lr] };
      v2f b11 = { sW[r2 + 16 + lr], sW[r2 + HH1 + 16 + lr] };
      acc0 = wmma4(aA, b00, acc0);
      acc1 = wmma4(aA, b01, acc1);
      acc0 = wmma4(aB, b10, acc0);
      acc1 = wmma4(aB, b11, acc1);
    }
    #pragma unroll
    for (int r = 0; r < 8; ++r) {          // C/D layout: M = r + lhi*8, N = lr
      const int m    = mt * 16 + r + lhi * 8;
      const int base = (m >> 1) * PSTR + (m & 1);
      supP[base + lr * 2]        = acc0[r];
      supP[base + (lr + 16) * 2] = acc1[r];
    }
  }
  __syncthreads();

  // ---- Phase 2: outs = adj @ sup via WMMA (adj streamed once) ----
  const float* ag = adj + (size_t)bw * NNN * NNN;
  float* og = outs + (size_t)bw * NNN * HH1;
  for (int mi = 0; mi < 4; ++mi) {
    const int mt  = wave * 4 + mi;
    const int row = mt * 16 + lr;
    const float* ap = ag + (size_t)row * NNN + lhi * 4;
    v8f acc0 = {}; v8f acc1 = {};
    #pragma unroll 8
    for (int k = 0; k < NNN; k += 8) {
      v4f a4 = *(const v4f*)(ap + k);
      v2f aA = {a4.x, a4.y};
      v2f aB = {a4.z, a4.w};
      // B fragments: one ds_load_b64 each, conflict-free paired layout
      const float* bp = &supP[((k >> 1) + lhi * 2) * PSTR + lr * 2];
      v2f b00 = *(const v2f*)(bp);
      v2f b01 = *(const v2f*)(bp + 32);          // col + 16
      v2f b10 = *(const v2f*)(bp + PSTR);        // K-pair + 1
      v2f b11 = *(const v2f*)(bp + PSTR + 32);
      acc0 = wmma4(aA, b00, acc0);   // A fragment reused for both N-tiles
      acc1 = wmma4(aA, b01, acc1);
      acc0 = wmma4(aB, b10, acc0);
      acc1 = wmma4(aB, b11, acc1);
    }
    #pragma unroll
    for (int r = 0; r < 8; ++r) {
      const int m = mt * 16 + r + lhi * 8;
      og[(size_t)m * HH1 + lr]      = acc0[r];
      og[(size_t)m * HH1 + 16 + lr] = acc1[r];
    }
  }
}

// ---------------------------------------------------------------------------
// Kernel 2: per node n. Cross-window L2 norm, per-node GEMM with W_inter[n],
// virtual fully-connected window graph (sum minus self), second L2 norm.
// 128 threads: t -> (w = t/32, h = t%32); loop over b.
// ---------------------------------------------------------------------------
__global__ __launch_bounds__(128)
void gcn_inter_kernel(const float* __restrict__ outs,
                      const float* __restrict__ Winter,
                      float* __restrict__ out)
{
  __shared__ float sWi[HH1 * HH2];   // 4 KB: W_inter[n]
  __shared__ float buf[WW * HH1];    // normalized features
  __shared__ float s2l[WW * HH2];    // support2

  const int n   = blockIdx.x;
  const int tid = threadIdx.x;
  const int w   = tid >> 5;          // window (also output window)
  const int h   = tid & 31;          // feature index (h and k)

  const float* Wg = Winter + (size_t)n * HH1 * HH2;
  for (int i = tid; i < HH1 * HH2; i += 128) sWi[i] = Wg[i];
  __syncthreads();

  for (int b = 0; b < BB; ++b) {
    const size_t base = (((size_t)b * WW + w) * NNN + n) * HH1 + h;
    float v = outs[base];
    buf[w * HH1 + h] = v;
    __syncthreads();
    // L2 norm across the 4 windows for this (n,h)
    float ss = 0.f;
    #pragma unroll
    for (int vv = 0; vv < WW; ++vv) { float t = buf[vv * HH1 + h]; ss += t * t; }
    float vn = v / fmaxf(sqrtf(ss), 1e-12f);
    __syncthreads();
    buf[w * HH1 + h] = vn;
    __syncthreads();
    // support2[w, k=h] = sum_hh buf[w,hh] * W_inter[n,hh,h]
    float s2 = 0.f;
    #pragma unroll
    for (int hh = 0; hh < HH1; ++hh)
      s2 = fmaf(buf[w * HH1 + hh], sWi[hh * HH2 + h], s2);
    s2l[w * HH2 + h] = s2;
    __syncthreads();
    // virtual graph: out[v] = (sum_w s2) - s2[v]; then L2 norm over v
    float tot = s2l[h] + s2l[HH2 + h] + s2l[2 * HH2 + h] + s2l[3 * HH2 + h];
    float o = tot - s2;
    float nn = 0.f;
    #pragma unroll
    for (int vv = 0; vv < WW; ++vv) { float t = tot - s2l[vv * HH2 + h]; nn += t * t; }
    out[base] = o / fmaxf(sqrtf(nn), 1e-12f);
    __syncthreads();   // protect LDS before next b iteration
  }
}

extern "C" void kernel_launch(void* const* d_in, const int* in_sizes, int n_in,
                              void* d_out, int out_size, void* d_ws, size_t ws_size,
                              hipStream_t stream) {
  const float* x      = (const float*)d_in[0];  // [B,W,N,F]
  const float* adj    = (const float*)d_in[1];  // [B,W,N,N]
  const float* Wintra = (const float*)d_in[2];  // [W,F,H1]
  const float* Winter = (const float*)d_in[3];  // [N,H1,H2]
  float* out = (float*)d_out;                   // [B,W,N,H2]
  float* outs_ws = (float*)d_ws;                // [B*W, N, H1] = 32 MB

  gcn_intra_kernel<<<BB * WW, 256, 0, stream>>>(x, adj, Wintra, outs_ws);
  gcn_inter_kernel<<<NNN, 128, 0, stream>>>(outs_ws, Winter, out);
}